// SpareNetGenerator_25211458027804
// MI455X (gfx1250) — compile-verified
//
#include <hip/hip_runtime.h>

typedef __attribute__((ext_vector_type(2))) float v2f;
typedef __attribute__((ext_vector_type(8))) float v8f;

#define B_    16
#define C_    64
#define N_    2048
#define K_    20
#define TM    16              // query rows per workgroup (one WMMA M tile)
#define NT    (N_ / TM)       // 128 column tiles
#define NWAVE 8
#define PAD   68              // LDS row pitch for A panel (bank-conflict pad)

// LDS layout (dynamic):
//   As[TM*PAD]           query panel, row-major with pad   (4352 B)
//   dist[TM*N_]          score matrix for this row tile    (131072 B)
//   topIdx[TM*K_]        selected neighbor indices         (1280 B)
#define SMEM_BYTES ((size_t)(TM * PAD + TM * N_) * sizeof(float) + (size_t)(TM * K_) * sizeof(int))

__global__ __launch_bounds__(256) void dgcnn_graph_feature(const float* __restrict__ x,
                                                           float* __restrict__ out) {
  extern __shared__ float smem[];
  float* As     = smem;                         // TM * PAD
  float* dist   = smem + TM * PAD;              // TM * N_
  int*   topIdx = (int*)(dist + TM * N_);       // TM * K_

  const int tid  = threadIdx.x;
  const int wave = tid >> 5;
  const int lane = tid & 31;
  const int hi   = lane >> 4;   // half-wave select
  const int L    = lane & 15;

  const int b  = blockIdx.x / NT;
  const int r0 = (blockIdx.x % NT) * TM;

  const float* __restrict__ xb = x + (size_t)b * C_ * N_;

  // ---- Stage query panel As[m][c] = x[b][c][r0+m] (coalesced over m) ----
  for (int e = tid; e < TM * C_; e += 256) {
    const int m = e & (TM - 1);
    const int c = e >> 4;
    As[m * PAD + c] = xb[(size_t)c * N_ + (r0 + m)];
  }
  __syncthreads();

  // ---- Score tiles: 2*<x_m, x_n> - |x_n|^2 via V_WMMA_F32_16X16X4_F32 ----
  for (int ct = wave; ct < NT; ct += NWAVE) {
    const int n0 = ct * TM;
    v8f acc = {0.f, 0.f, 0.f, 0.f, 0.f, 0.f, 0.f, 0.f};
    float sq = 0.f;   // partial |x_n|^2 for column n0+L (half the channels)

#pragma unroll
    for (int t = 0; t < 16; ++t) {
      const int k0 = 4 * t + 2 * hi;
      v2f a, bb;
      // A 16x4 f32 layout: lanes 0-15 hold K={4t,4t+1}, lanes 16-31 K={4t+2,4t+3}
      a.x = As[L * PAD + k0];
      a.y = As[L * PAD + k0 + 1];
      // B 4x16 f32 layout mirrors A; N = lane & 15
      bb.x = xb[(size_t)k0 * N_ + n0 + L];
      bb.y = xb[(size_t)(k0 + 1) * N_ + n0 + L];
      sq += bb.x * bb.x + bb.y * bb.y;
      acc = __builtin_amdgcn_wmma_f32_16x16x4_f32(false, a, false, bb,
                                                  (short)0, acc, false, false);
    }
    // combine the two half-wave channel halves of |x_n|^2
    sq += __shfl_xor(sq, 16);

    // C/D layout: VGPR r -> M = 8*hi + r, N = lane & 15
#pragma unroll
    for (int r = 0; r < 8; ++r) {
      dist[(hi * 8 + r) * N_ + n0 + L] = 2.0f * acc[r] - sq;
    }
  }
  __syncthreads();

  // ---- Top-20 per row: 20 wave-parallel argmax passes with invalidation ----
  // Tie-break toward lower index to match jax.lax.top_k stability.
  for (int m = wave * 2; m < wave * 2 + 2; ++m) {
    float* __restrict__ D = dist + m * N_;
    for (int j = 0; j < K_; ++j) {
      float best = -3.402823466e38f;
      int bestN  = -1;
      for (int n = lane; n < N_; n += 32) {
        const float s = D[n];
        if (s > best) { best = s; bestN = n; }   // strict > keeps lowest n on ties
      }
#pragma unroll
      for (int off = 16; off > 0; off >>= 1) {
        const float ob = __shfl_xor(best, off);
        const int   on = __shfl_xor(bestN, off);
        if (ob > best || (ob == best && (unsigned)on < (unsigned)bestN)) {
          best = ob; bestN = on;
        }
      }
      if (lane == 0) {
        topIdx[m * K_ + j] = bestN;
        D[bestN] = -__builtin_inff();            // invalidate for next pass
      }
      __builtin_amdgcn_wave_barrier();           // keep DS write->read ordered
    }
  }
  __syncthreads();

  // ---- Emit output: out[b][c][n][j] = x_nbr[c]-x_n[c] ; out[b][64+c][n][j] = x_n[c] ----
  float* __restrict__ outB = out + (size_t)b * (2 * C_) * N_ * K_;
  for (int m = wave * 2; m < wave * 2 + 2; ++m) {
    const int nRow = r0 + m;
    for (int e = lane; e < C_ * K_; e += 32) {
      const int c  = e / K_;
      const int j  = e - c * K_;
      const int nj = topIdx[m * K_ + j];
      const float center = As[m * PAD + c];
      const float neigh  = xb[(size_t)c * N_ + nj];
      outB[((size_t)c         * N_ + nRow) * K_ + j] = neigh - center;
      outB[((size_t)(c + C_)  * N_ + nRow) * K_ + j] = center;
    }
  }
}

extern "C" void kernel_launch(void* const* d_in, const int* in_sizes, int n_in,
                              void* d_out, int out_size, void* d_ws, size_t ws_size,
                              hipStream_t stream) {
  (void)in_sizes; (void)n_in; (void)d_ws; (void)ws_size; (void)out_size;
  const float* x = (const float*)d_in[0];   // (16, 64, 2048) f32
  float* out = (float*)d_out;               // (16, 128, 2048, 20) f32
  dgcnn_graph_feature<<<dim3(B_ * NT), dim3(256), SMEM_BYTES, stream>>>(x, out);
}